// GraphConvolutionalNetwork_24816321036837
// MI455X (gfx1250) — compile-verified
//
#include <hip/hip_runtime.h>
#include <hip/hip_bf16.h>
#include <math.h>

// ---------------------------------------------------------------------------
// GCN pipeline for MI455X (gfx1250, wave32).
// Dense [N,32]@[32,32] GEMMs use V_WMMA_F32_16X16X4_F32 (one 16-row tile per
// wave, K=32 => 8 wmma steps per 16-col half, 16 wmma per wave).
// Edge aggregation uses native GLOBAL_ATOMIC_ADD_F32 (non-returning, L2
// fire-and-forget) via inline asm so the compiler cannot fall back to a CAS
// loop for these coarse-grained workspace buffers.
// ---------------------------------------------------------------------------

#define DFEAT 32

typedef float v2f __attribute__((ext_vector_type(2)));
typedef float v8f __attribute__((ext_vector_type(8)));

static __device__ __forceinline__ float relu_f(float x) { return fmaxf(x, 0.0f); }

// Non-returning hardware FP32 atomic add (ISA 15.18.3 op 86, GLOBAL_ATOMIC_ADD_F32).
// TH[0]=0 (no return) => tracked on STOREcnt; kernel-end implicit wait covers it.
static __device__ __forceinline__ void atomic_add_f32(float* p, float v) {
  asm volatile("global_atomic_add_f32 %0, %1, off" :: "v"(p), "v"(v) : "memory");
}

// ---------------- elementwise / setup kernels ------------------------------

__global__ void k_zero(float* __restrict__ p, int n) {
  int i = blockIdx.x * blockDim.x + threadIdx.x;
  if (i < n) p[i] = 0.0f;
}

// h0[n,d] = relu(x[n]*W0[d] + b0[d])   (x is [N,1] so layer-0 is outer product)
__global__ void k_pre0(const float* __restrict__ x, const float* __restrict__ W0,
                       const float* __restrict__ b0, float* __restrict__ h, int n) {
  int i = blockIdx.x * blockDim.x + threadIdx.x;
  if (i >= n * DFEAT) return;
  int node = i >> 5, d = i & 31;
  h[i] = relu_f(fmaf(x[node], W0[d], b0[d]));
}

__global__ void k_deg_accum(const int* __restrict__ col, const float* __restrict__ w,
                            float* __restrict__ deg, int E) {
  int e = blockIdx.x * blockDim.x + threadIdx.x;
  if (e < E) atomic_add_f32(&deg[col[e]], w[e]);
}

// deg += 1 (self loop); dinv = rsqrt(deg); invdeg = 1/deg
__global__ void k_deg_finalize(const float* __restrict__ deg, float* __restrict__ dinv,
                               float* __restrict__ invdeg, int n) {
  int i = blockIdx.x * blockDim.x + threadIdx.x;
  if (i >= n) return;
  float d = deg[i] + 1.0f;
  dinv[i]   = rsqrtf(d);
  invdeg[i] = 1.0f / d;
}

__global__ void k_norm(const int* __restrict__ row, const int* __restrict__ col,
                       const float* __restrict__ w, const float* __restrict__ dinv,
                       float* __restrict__ norm, int E) {
  int e = blockIdx.x * blockDim.x + threadIdx.x;
  if (e < E) norm[e] = dinv[row[e]] * w[e] * dinv[col[e]];
}

// ---------------- WMMA GEMM: C[N,32] = act(A[N,32] @ W[32,32] + bias) -------
// 128 threads = 4 waves per block; each wave owns one 16-row tile.
// Fragment layouts per CDNA5 ISA 7.12.2 (32-bit A 16x4 / C 16x16 tables).
__global__ void k_gemm32_wmma(const float* __restrict__ A, const float* __restrict__ W,
                              const float* __restrict__ bias, float* __restrict__ C,
                              int nrows, int do_relu) {
  __shared__ float sW[DFEAT * DFEAT];
  __shared__ float sB[DFEAT];
  int tid = threadIdx.x;
  for (int i = tid; i < DFEAT * DFEAT; i += blockDim.x) sW[i] = W[i];
  if (tid < DFEAT) sB[tid] = bias ? bias[tid] : 0.0f;
  __syncthreads();

  int wave   = tid >> 5;
  int lane   = tid & 31;
  int tile   = blockIdx.x * 4 + wave;        // 16-row output tile
  int row0   = tile * 16;
  if (row0 >= nrows) return;                 // wave-uniform guard (EXEC all-1s at wmma)
  int half   = lane >> 4;                    // 0: K pair 0,1   1: K pair 2,3
  int lanelo = lane & 15;

  const float* Arow = A + (size_t)(row0 + lanelo) * DFEAT;

  v8f c0 = {};  // columns 0..15
  v8f c1 = {};  // columns 16..31
#pragma unroll
  for (int kk = 0; kk < 8; ++kk) {
    int ka = kk * 4 + half * 2;              // this lane's two K indices
    v2f a;  a.x  = Arow[ka];                 a.y  = Arow[ka + 1];
    v2f b0; b0.x = sW[ka * DFEAT + lanelo];  b0.y = sW[(ka + 1) * DFEAT + lanelo];
    v2f b1; b1.x = sW[ka * DFEAT + 16 + lanelo];
            b1.y = sW[(ka + 1) * DFEAT + 16 + lanelo];
    c0 = __builtin_amdgcn_wmma_f32_16x16x4_f32(false, a, false, b0, (short)0, c0, false, false);
    c1 = __builtin_amdgcn_wmma_f32_16x16x4_f32(false, a, false, b1, (short)0, c1, false, false);
  }

#pragma unroll
  for (int r = 0; r < 8; ++r) {
    int orow = row0 + half * 8 + r;          // C layout: VGPR r -> M = r + 8*half
    float v0 = c0[r] + sB[lanelo];
    float v1 = c1[r] + sB[16 + lanelo];
    if (do_relu) { v0 = relu_f(v0); v1 = relu_f(v1); }
    C[(size_t)orow * DFEAT + lanelo]      = v0;
    C[(size_t)orow * DFEAT + 16 + lanelo] = v1;
  }
}

// ---------------- GCN aggregation ------------------------------------------

// agg[n,:] = xw[n,:] / deg[n]   (self-loop message)
__global__ void k_selfloop(const float* __restrict__ xw, const float* __restrict__ invdeg,
                           float* __restrict__ agg, int n) {
  int i = blockIdx.x * blockDim.x + threadIdx.x;
  if (i >= n * DFEAT) return;
  agg[i] = xw[i] * invdeg[i >> 5];
}

// 8 threads per edge, 4 features each: agg[col] += norm[e] * xw[row]
__global__ void k_scatter(const int* __restrict__ row, const int* __restrict__ col,
                          const float* __restrict__ norm, const float* __restrict__ xw,
                          float* __restrict__ agg, int E) {
  int idx = blockIdx.x * blockDim.x + threadIdx.x;
  int e = idx >> 3;
  if (e >= E) return;
  int q = (idx & 7) << 2;
  float nm = norm[e];
  float4 v = *(const float4*)(xw + (size_t)row[e] * DFEAT + q);
  float* dst = agg + (size_t)col[e] * DFEAT + q;
  atomic_add_f32(dst + 0, nm * v.x);
  atomic_add_f32(dst + 1, nm * v.y);
  atomic_add_f32(dst + 2, nm * v.z);
  atomic_add_f32(dst + 3, nm * v.w);
}

__global__ void k_biasrelu(const float* __restrict__ agg, const float* __restrict__ b,
                           float* __restrict__ h, int n) {
  int i = blockIdx.x * blockDim.x + threadIdx.x;
  if (i >= n * DFEAT) return;
  h[i] = relu_f(agg[i] + b[i & 31]);
}

// ---------------- pooling + post-MLP ---------------------------------------

__global__ void k_pool(const float* __restrict__ h, const int* __restrict__ batch,
                       float* __restrict__ sums, float* __restrict__ cnts, int n) {
  int i = blockIdx.x * blockDim.x + threadIdx.x;
  if (i >= n * DFEAT) return;
  int node = i >> 5, d = i & 31;
  int g = batch[node];
  atomic_add_f32(&sums[(size_t)g * DFEAT + d], h[i]);
  if (d == 0) atomic_add_f32(&cnts[g], 1.0f);
}

__global__ void k_post(const float* __restrict__ sums, const float* __restrict__ cnts,
                       const float* __restrict__ W0, const float* __restrict__ b0,
                       const float* __restrict__ W1, const float* __restrict__ b1,
                       float* __restrict__ out, int ngraphs) {
  int g = blockIdx.x * blockDim.x + threadIdx.x;
  if (g >= ngraphs) return;
  float inv = 1.0f / fmaxf(cnts[g], 1.0f);
  float gv[DFEAT];
#pragma unroll
  for (int d = 0; d < DFEAT; ++d) gv[d] = sums[(size_t)g * DFEAT + d] * inv;
  float acc = b1[0];
  for (int j = 0; j < DFEAT; ++j) {
    float a = b0[j];
#pragma unroll
    for (int d = 0; d < DFEAT; ++d) a = fmaf(gv[d], W0[d * DFEAT + j], a);
    acc = fmaf(relu_f(a), W1[j], acc);
  }
  out[g] = 1.0f / (1.0f + __expf(-acc));
}

// ---------------------------------------------------------------------------

static inline int cdiv(int a, int b) { return (a + b - 1) / b; }

extern "C" void kernel_launch(void* const* d_in, const int* in_sizes, int n_in,
                              void* d_out, int out_size, void* d_ws, size_t ws_size,
                              hipStream_t stream) {
  const float* x       = (const float*)d_in[0];
  const int*   eidx    = (const int*)  d_in[1];
  const float* ew      = (const float*)d_in[2];
  const int*   batch   = (const int*)  d_in[3];
  const float* W_pre0  = (const float*)d_in[4];
  const float* b_pre0  = (const float*)d_in[5];
  const float* W_pre1  = (const float*)d_in[6];
  const float* b_pre1  = (const float*)d_in[7];
  const float* W_g[3]  = {(const float*)d_in[8],  (const float*)d_in[10], (const float*)d_in[12]};
  const float* b_g[3]  = {(const float*)d_in[9],  (const float*)d_in[11], (const float*)d_in[13]};
  const float* W_post0 = (const float*)d_in[14];
  const float* b_post0 = (const float*)d_in[15];
  const float* W_post1 = (const float*)d_in[16];
  const float* b_post1 = (const float*)d_in[17];
  float* out = (float*)d_out;

  const int N = in_sizes[0];          // 160000 nodes (x is [N,1])
  const int E = in_sizes[2];          // 2560000 edges
  const int G = out_size;             // 1024 graphs
  const int* row = eidx;              // edge_indices[0,:]
  const int* col = eidx + E;          // edge_indices[1,:]

  // ---- carve workspace (256B aligned chunks) ----
  char* ws = (char*)d_ws;
  size_t off = 0;
  auto carve = [&](size_t bytes) -> float* {
    float* p = (float*)(ws + off);
    off += (bytes + 255) & ~(size_t)255;
    return p;
  };
  float* hA     = carve((size_t)N * DFEAT * 4);   // gemm out / xw
  float* hB     = carve((size_t)N * DFEAT * 4);   // current features h
  float* hC     = carve((size_t)N * DFEAT * 4);   // agg
  float* norm   = carve((size_t)E * 4);
  float* deg    = carve((size_t)N * 4);
  float* dinv   = carve((size_t)N * 4);
  float* invdeg = carve((size_t)N * 4);
  float* sums   = carve((size_t)G * DFEAT * 4);
  float* cnts   = carve((size_t)G * 4);
  (void)ws_size;
  (void)cnts; // contiguous after sums; zeroed/used via sums base arithmetic below

  const int TPB = 256;
  const int nElem  = N * DFEAT;
  const int gElem  = cdiv(nElem, TPB);
  const int gNodes = cdiv(N, TPB);
  const int gEdges = cdiv(E, TPB);
  const int gGemm  = cdiv(cdiv(N, 16), 4);        // 4 waves (16-row tiles) per block

  // ---- degree / normalization (shared by all 3 GCN layers) ----
  k_zero<<<gNodes, TPB, 0, stream>>>(deg, N);
  k_zero<<<cdiv(G * DFEAT + G, TPB), TPB, 0, stream>>>(sums, G * DFEAT + G); // sums+cnts contiguous
  k_deg_accum<<<gEdges, TPB, 0, stream>>>(col, ew, deg, E);
  k_deg_finalize<<<gNodes, TPB, 0, stream>>>(deg, dinv, invdeg, N);
  k_norm<<<gEdges, TPB, 0, stream>>>(row, col, ew, dinv, norm, E);

  // ---- pre-MLP ----
  k_pre0<<<gElem, TPB, 0, stream>>>(x, W_pre0, b_pre0, hA, N);
  k_gemm32_wmma<<<gGemm, 128, 0, stream>>>(hA, W_pre1, b_pre1, hB, N, 1);

  // ---- 3x GCNConv + ReLU ----
  for (int l = 0; l < 3; ++l) {
    k_gemm32_wmma<<<gGemm, 128, 0, stream>>>(hB, W_g[l], nullptr, hA, N, 0); // xw
    k_selfloop<<<gElem, TPB, 0, stream>>>(hA, invdeg, hC, N);
    k_scatter<<<cdiv(E * 8, TPB), TPB, 0, stream>>>(row, col, norm, hA, hC, E);
    k_biasrelu<<<gElem, TPB, 0, stream>>>(hC, b_g[l], hB, N);
  }

  // ---- mean pool + post-MLP ----
  k_pool<<<gElem, TPB, 0, stream>>>(hB, batch, sums, cnts, N);
  k_post<<<cdiv(G, TPB), TPB, 0, stream>>>(sums, cnts, W_post0, b_post0, W_post1, b_post1,
                                           out, G);
}